// GoedeckerNet_50319836840322
// MI455X (gfx1250) — compile-verified
//
#include <hip/hip_runtime.h>
#include <hip/hip_bf16.h>
#include <math.h>
#include <stdint.h>

#define NBATCH 16
#define NA 1024
#define NDESC 128
#define NHID 64
#define NELEM 2
#define NT 64          // number of 16x16 block tiles per dim (1024/16)
#define SQRT2OPI 0.7978845608028654f

typedef __attribute__((ext_vector_type(2))) float v2f;
typedef __attribute__((ext_vector_type(8))) float v8f;

// D(16x16,f32) = A(16x4,f32) x B(4x16,f32) + C   -- CDNA5 V_WMMA_F32_16X16X4_F32
__device__ __forceinline__ v8f wmma4(v2f a, v2f b, v8f c) {
  return __builtin_amdgcn_wmma_f32_16x16x4_f32(
      /*neg_a=*/false, a, /*neg_b=*/false, b,
      /*c_mod=*/(short)0, c, /*reuse_a=*/false, /*reuse_b=*/false);
}

__device__ __forceinline__ float block_reduce256(float v, float* red, int tid) {
  red[tid] = v;
  __syncthreads();
  for (int s = 128; s > 0; s >>= 1) {
    if (tid < s) red[tid] += red[tid + s];
    __syncthreads();
  }
  float r = red[0];
  __syncthreads();
  return r;
}

// ---------------------------------------------------------------------------
// Kernel 1: per-atom MLP -> chi.  One wave = 16 atoms; layers 1&2 via WMMA f32.
// ---------------------------------------------------------------------------
__global__ __launch_bounds__(128) void mlp_chi_kernel(
    const float* __restrict__ desc, const int* __restrict__ atnum,
    const float* __restrict__ W1, const float* __restrict__ b1,
    const float* __restrict__ W2, const float* __restrict__ b2,
    const float* __restrict__ W3, const float* __restrict__ b3,
    float* __restrict__ chi) {
  __shared__ float hbuf[4][16 * NHID];
  __shared__ float h2buf[4][16 * NHID];
  const int lane = threadIdx.x & 31;
  const int wv = threadIdx.x >> 5;
  const int group = blockIdx.x * 4 + wv;     // 0 .. NBATCH*NA/16-1
  const int b = group >> 6;                  // 64 groups per batch
  const int row0 = (group & 63) * 16;
  const int m = lane & 15;                   // row (M) or col (N) index
  const int hh = lane >> 4;                  // half-wave select
  const float* drow = desc + ((size_t)b * NA + row0) * NDESC;

  for (int e = 0; e < NELEM; ++e) {
    const float* w1e = W1 + (size_t)e * NDESC * NHID;
    // layer 1: (16 x 128) * (128 x 64)
    for (int t = 0; t < 4; ++t) {
      v8f acc = {};
      for (int kk = 0; kk < NDESC / 4; ++kk) {
        const int kb = kk * 4 + 2 * hh;
        v2f a, bb;
        a.x = drow[m * NDESC + kb];
        a.y = drow[m * NDESC + kb + 1];
        bb.x = w1e[kb * NHID + t * 16 + m];
        bb.y = w1e[(kb + 1) * NHID + t * 16 + m];
        acc = wmma4(a, bb, acc);
      }
      const float bias = b1[e * NHID + t * 16 + m];
      for (int v = 0; v < 8; ++v)
        hbuf[wv][(v + 8 * hh) * NHID + t * 16 + m] = tanhf(acc[v] + bias);
    }
    __syncthreads();
    // layer 2: (16 x 64) * (64 x 64)
    const float* w2e = W2 + (size_t)e * NHID * NHID;
    for (int t = 0; t < 4; ++t) {
      v8f acc = {};
      for (int kk = 0; kk < NHID / 4; ++kk) {
        const int kb = kk * 4 + 2 * hh;
        v2f a, bb;
        a.x = hbuf[wv][m * NHID + kb];
        a.y = hbuf[wv][m * NHID + kb + 1];
        bb.x = w2e[kb * NHID + t * 16 + m];
        bb.y = w2e[(kb + 1) * NHID + t * 16 + m];
        acc = wmma4(a, bb, acc);
      }
      const float bias = b2[e * NHID + t * 16 + m];
      for (int v = 0; v < 8; ++v)
        h2buf[wv][(v + 8 * hh) * NHID + t * 16 + m] = tanhf(acc[v] + bias);
    }
    __syncthreads();
    // layer 3 (vector) + per-element selection
    if (lane < 16) {
      float acc = 0.f;
      for (int k = 0; k < NHID; ++k)
        acc += h2buf[wv][m * NHID + k] * W3[e * NHID + k];
      const float en = tanhf(acc + b3[e]);
      const int ga = b * NA + row0 + m;
      if (atnum[ga] == e) chi[ga] = en;
    }
    __syncthreads();
  }
}

// ---------------------------------------------------------------------------
// Kernel 2: build the hardness/Coulomb matrix A_ (16 x 1024 x 1024)
// ---------------------------------------------------------------------------
__global__ __launch_bounds__(256) void build_A_kernel(
    const float* __restrict__ pos, const int* __restrict__ atnum,
    const float* __restrict__ gw, const float* __restrict__ hard,
    float* __restrict__ A) {
  const size_t idx = (size_t)blockIdx.x * 256 + threadIdx.x;
  const int b = (int)(idx >> 20);
  const int ij = (int)(idx & ((1u << 20) - 1));
  const int i = ij >> 10;
  const int j = ij & 1023;
  const int zi = atnum[b * NA + i];
  float val;
  if (i == j) {
    val = hard[zi] + SQRT2OPI / gw[zi];
  } else {
    const int zj = atnum[b * NA + j];
    const float* pi = pos + ((size_t)b * NA + i) * 3;
    const float* pj = pos + ((size_t)b * NA + j) * 3;
    const float dx = pj[0] - pi[0];
    const float dy = pj[1] - pi[1];
    const float dz = pj[2] - pi[2];
    const float r = sqrtf(dx * dx + dy * dy + dz * dz);
    const float ai = gw[zi], aj = gw[zj];
    const float gamma = rsqrtf(ai * ai + aj * aj);
    val = erff(gamma * r) / r;
  }
  A[idx] = val;
}

// ---------------------------------------------------------------------------
// Kernel 3 (per LU step k): invert diagonal block (Gauss-Jordan in LDS),
// store Dinv, then panel Lneg[i,k] = -A[i,k] * Dinv via WMMA (overwrites A).
// One workgroup (8 waves) per batch.
// ---------------------------------------------------------------------------
__global__ __launch_bounds__(256) void panel_kernel(
    float* __restrict__ A, float* __restrict__ Dinv, int k) {
  __shared__ float aug[16][33];
  __shared__ float dinvS[16][16];
  const int b = blockIdx.x;
  float* Ab = A + (size_t)b * NA * NA;
  const int tid = threadIdx.x;

  // load [A_kk | I] into aug
  for (int e = tid; e < 16 * 32; e += 256) {
    const int r = e >> 5, c = e & 31;
    aug[r][c] = (c < 16) ? Ab[(size_t)(k * 16 + r) * NA + k * 16 + c]
                         : ((c - 16 == r) ? 1.f : 0.f);
  }
  __syncthreads();

  // Gauss-Jordan (no pivoting; matrix is SPD-dominated)
  for (int p = 0; p < 16; ++p) {
    const float pinv = 1.0f / aug[p][p];
    __syncthreads();
    if (tid < 32) aug[p][tid] *= pinv;
    __syncthreads();
    const int e0 = tid, e1 = tid + 256;
    const int r0 = e0 >> 5, c0 = e0 & 31;
    const int r1 = e1 >> 5, c1 = e1 & 31;
    const float f0 = aug[r0][p], f1 = aug[r1][p];
    const float u0 = aug[p][c0], u1 = aug[p][c1];
    __syncthreads();
    if (r0 != p) aug[r0][c0] -= f0 * u0;
    if (r1 != p) aug[r1][c1] -= f1 * u1;
    __syncthreads();
  }

  // stash inverse (LDS + global)
  {
    const int r = tid >> 4, c = tid & 15;
    const float v = aug[r][16 + c];
    dinvS[r][c] = v;
    Dinv[((size_t)b * NT + k) * 256 + r * 16 + c] = v;
  }
  __syncthreads();

  // panel: A[i,k] <- -(A[i,k] * Dinv), one block-row per wave round-robin
  const int wv = tid >> 5, lane = tid & 31;
  const int m = lane & 15, hh = lane >> 4;
  for (int i = k + 1 + wv; i < NT; i += 8) {
    v8f acc = {};
    for (int kk = 0; kk < 4; ++kk) {
      const int kb = kk * 4 + 2 * hh;
      v2f a, bb;
      a.x = Ab[(size_t)(i * 16 + m) * NA + k * 16 + kb];
      a.y = Ab[(size_t)(i * 16 + m) * NA + k * 16 + kb + 1];
      bb.x = dinvS[kb][m];
      bb.y = dinvS[kb + 1][m];
      acc = wmma4(a, bb, acc);
    }
    for (int v = 0; v < 8; ++v)
      Ab[(size_t)(i * 16 + v + 8 * hh) * NA + k * 16 + m] = -acc[v];
  }
}

// ---------------------------------------------------------------------------
// Kernel 4 (per LU step k): trailing update A[i,j] += Lneg[i,k]*U[k,j].
// One workgroup (8 waves) per trailing block-row i.  The 64KB pivot block-row
// U[k,*] is staged into LDS once per block with async global->LDS b128 copies
// (ASYNCcnt), then shared by all 8 waves; L fragments live in VGPRs across
// the whole j sweep.
// ---------------------------------------------------------------------------
__global__ __launch_bounds__(256) void trailing_kernel(float* __restrict__ A, int k) {
  __shared__ float ldsU[16 * NA];   // 64 KB: rows k*16 .. k*16+15 of Ab
  const int b = blockIdx.y;
  float* Ab = A + (size_t)b * NA * NA;
  const int tid = threadIdx.x;

  // async-stage the (contiguous) pivot block-row into LDS
  {
    const uint64_t gbase = (uint64_t)(uintptr_t)(Ab + (size_t)k * 16 * NA);
    const uint32_t lbase =
        (uint32_t)(uintptr_t)(__attribute__((address_space(3))) char*)ldsU;
    for (int it = 0; it < 16; ++it) {
      const uint32_t voff = (uint32_t)(it * 256 + tid) * 16u;  // 16B per lane
      const uint32_t laddr = lbase + voff;
      asm volatile("global_load_async_to_lds_b128 %0, %1, %2"
                   :: "v"(laddr), "v"(voff), "s"(gbase) : "memory");
    }
    asm volatile("s_wait_asynccnt 0x0" ::: "memory");
  }
  __syncthreads();

  const int wv = tid >> 5, lane = tid & 31;
  const int m = lane & 15, hh = lane >> 4;
  const int i = k + 1 + blockIdx.x;

  // L fragments: loaded once, reused for the whole j sweep
  v2f aL[4];
#pragma unroll
  for (int kk = 0; kk < 4; ++kk) {
    const int kb = kk * 4 + 2 * hh;
    aL[kk].x = Ab[(size_t)(i * 16 + m) * NA + k * 16 + kb];
    aL[kk].y = Ab[(size_t)(i * 16 + m) * NA + k * 16 + kb + 1];
  }

  for (int j = k + 1 + wv; j < NT; j += 8) {
    float* C = Ab + (size_t)(i * 16) * NA + j * 16;
    v8f acc;
#pragma unroll
    for (int v = 0; v < 8; ++v) acc[v] = C[(size_t)(v + 8 * hh) * NA + m];
#pragma unroll
    for (int kk = 0; kk < 4; ++kk) {
      const int kb = kk * 4 + 2 * hh;
      v2f bb;
      bb.x = ldsU[kb * NA + j * 16 + m];
      bb.y = ldsU[(kb + 1) * NA + j * 16 + m];
      acc = wmma4(aL[kk], bb, acc);
    }
#pragma unroll
    for (int v = 0; v < 8; ++v) C[(size_t)(v + 8 * hh) * NA + m] = acc[v];
  }
}

// ---------------------------------------------------------------------------
// Kernel 5: per-batch triangular solves (2 RHS: chi and ones), Schur lambda,
// energy E = 1.5*chi.q - 0.5*lambda*q_tot.
// ---------------------------------------------------------------------------
__global__ __launch_bounds__(256) void solve_energy_kernel(
    const float* __restrict__ A, const float* __restrict__ Dinv,
    const float* __restrict__ chi, const float* __restrict__ qtot,
    float* __restrict__ out) {
  __shared__ float y[2][NA];
  __shared__ float tb[2][16];
  __shared__ float red[256];
  const int b = blockIdx.x;
  const float* Ab = A + (size_t)b * NA * NA;
  const float* chib = chi + b * NA;
  const int tid = threadIdx.x;

  for (int i2 = tid; i2 < NA; i2 += 256) {
    y[0][i2] = chib[i2];
    y[1][i2] = 1.0f;
  }
  __syncthreads();

  // forward: y_i += sum_{c<i} Lneg[i,c] * y_c   (Lneg stored => addition)
  for (int i = 1; i < NT; ++i) {
    if (tid < 32) {
      const int r = tid >> 4, m = tid & 15;
      const int row = i * 16 + m;
      float acc = y[r][row];
      const float* Lrow = Ab + (size_t)row * NA;
      for (int c = 0; c < i * 16; ++c) acc += Lrow[c] * y[r][c];
      y[r][row] = acc;
    }
    __syncthreads();
  }

  // backward: x_i = Dinv_i * (y_i - sum_{c>i} U[i,c] x_c)
  for (int i = NT - 1; i >= 0; --i) {
    if (tid < 32) {
      const int r = tid >> 4, m = tid & 15;
      const int row = i * 16 + m;
      float acc = y[r][row];
      const float* Urow = Ab + (size_t)row * NA;
      for (int c = (i + 1) * 16; c < NA; ++c) acc -= Urow[c] * y[r][c];
      tb[r][m] = acc;
    }
    __syncthreads();
    if (tid < 32) {
      const int r = tid >> 4, m = tid & 15;
      const float* Di = Dinv + ((size_t)b * NT + i) * 256;
      float acc = 0.f;
      for (int c = 0; c < 16; ++c) acc += Di[m * 16 + c] * tb[r][c];
      y[r][i * 16 + m] = acc;
    }
    __syncthreads();
  }

  // lambda and energy
  float p1 = 0.f, p2 = 0.f;
  for (int i2 = tid; i2 < NA; i2 += 256) {
    p1 += y[0][i2];
    p2 += y[1][i2];
  }
  const float s1 = block_reduce256(p1, red, tid);
  const float s2 = block_reduce256(p2, red, tid);
  const float qt = qtot[b];
  const float lam = (s1 - qt) / s2;

  float pe = 0.f;
  for (int i2 = tid; i2 < NA; i2 += 256) {
    const float q = y[0][i2] - lam * y[1][i2];
    pe += chib[i2] * q;
  }
  const float e1 = block_reduce256(pe, red, tid);
  if (tid == 0) out[b] = 1.5f * e1 - 0.5f * lam * qt;
}

// ---------------------------------------------------------------------------
extern "C" void kernel_launch(void* const* d_in, const int* in_sizes, int n_in,
                              void* d_out, int out_size, void* d_ws, size_t ws_size,
                              hipStream_t stream) {
  (void)in_sizes; (void)n_in; (void)out_size; (void)ws_size;
  const float* desc = (const float*)d_in[0];
  const float* pos  = (const float*)d_in[1];
  const int*   atn  = (const int*)d_in[2];
  // d_in[3] all_neighbors (identity) and d_in[4] all_mask (1-eye) are implicit
  const float* qtot = (const float*)d_in[5];
  const float* W1   = (const float*)d_in[6];
  const float* b1   = (const float*)d_in[7];
  const float* W2   = (const float*)d_in[8];
  const float* b2   = (const float*)d_in[9];
  const float* W3   = (const float*)d_in[10];
  const float* b3   = (const float*)d_in[11];
  const float* gw   = (const float*)d_in[12];
  const float* hard = (const float*)d_in[13];
  float* out = (float*)d_out;

  float* ws   = (float*)d_ws;
  float* A    = ws;                                   // 16*1024*1024 floats (64 MB)
  float* Dinv = A + (size_t)NBATCH * NA * NA;         // 16*64*256 floats (1 MB)
  float* chi  = Dinv + (size_t)NBATCH * NT * 256;     // 16*1024 floats

  mlp_chi_kernel<<<NBATCH * NA / 64, 128, 0, stream>>>(desc, atn, W1, b1, W2, b2,
                                                       W3, b3, chi);
  build_A_kernel<<<(NBATCH * NA * NA) / 256, 256, 0, stream>>>(pos, atn, gw, hard, A);

  for (int k = 0; k < NT; ++k) {
    panel_kernel<<<NBATCH, 256, 0, stream>>>(A, Dinv, k);
    const int rem = NT - 1 - k;
    if (rem > 0)
      trailing_kernel<<<dim3(rem, NBATCH), 256, 0, stream>>>(A, k);
  }

  solve_energy_kernel<<<NBATCH, 256, 0, stream>>>(A, Dinv, chi, qtot, out);
}